// HausdorffLoss_78640851189976
// MI455X (gfx1250) — compile-verified
//
#include <hip/hip_runtime.h>
#include <math.h>

// Soft Hausdorff loss, B=4, N=M=4096, D=3, fp32.
// One v_wmma_f32_16x16x4_f32 produces a full 16x16 squared-distance tile:
//   A = -2 * p1 (16x4, xyz0), B = p2 (4x16, xyz0), C = sq1[row] + sq2[col].
// Minima are tracked with a monotone uint encoding so float-min == umin:
//   branchless ds_min_u32 (column minima in LDS) + global_atomic_min_u32.

#define PTS   4096
#define BATCH 4
#define ALPHAF 10.0f
#define MSPLIT 2

typedef __attribute__((ext_vector_type(2))) float v2f;
typedef __attribute__((ext_vector_type(8))) float v8f;

// Monotone order-preserving float<->uint map: umin(enc(a),enc(b)) == enc(fmin(a,b))
__device__ __forceinline__ unsigned encMinF(float f) {
  unsigned b = __float_as_uint(f);
  return b ^ ((unsigned)((int)b >> 31) | 0x80000000u);
}
__device__ __forceinline__ float decMinF(unsigned u) {
  return __uint_as_float((u & 0x80000000u) ? (u ^ 0x80000000u) : ~u);
}

__global__ __launch_bounds__(256) void hd_init(unsigned* buf, int n) {
  int i = blockIdx.x * 256 + threadIdx.x;
  if (i < n) buf[i] = 0xFFFFFFFFu; // encoded +max (every slot is overwritten by real mins)
}

__global__ __launch_bounds__(256) void hd_main(const float* __restrict__ p1,
                                               const float* __restrict__ p2,
                                               unsigned* __restrict__ min_n,
                                               unsigned* __restrict__ min_m) {
  const int MT_PER_BLOCK = (PTS / 16) / MSPLIT; // 128 m-tiles per block
  const int MCOLS = MT_PER_BLOCK * 16;          // 2048 columns per block
  __shared__ float4   lds_p2q[MCOLS];           // {x, y, z, |p|^2}  (32 KB)
  __shared__ unsigned lds_colmin[MCOLS];        // encoded col minima (8 KB)
  __shared__ float    lds_sq1[128];

  int bid   = blockIdx.x;
  int b     = bid / ((PTS / 128) * MSPLIT);
  int rem   = bid % ((PTS / 128) * MSPLIT);
  int nblk  = rem / MSPLIT;
  int mpart = rem % MSPLIT;

  const float* P1 = p1 + (size_t)b * PTS * 3;
  const float* P2 = p2 + (size_t)b * PTS * 3;
  int nbase  = nblk * 128;
  int mstart = mpart * MCOLS;
  int tid = threadIdx.x;

  // Stage p2 slice as {x,y,z,sq2} quads; init LDS column minima
  for (int i = tid; i < MCOLS; i += 256) {
    float x = P2[(mstart + i) * 3 + 0];
    float y = P2[(mstart + i) * 3 + 1];
    float z = P2[(mstart + i) * 3 + 2];
    lds_p2q[i] = make_float4(x, y, z, x * x + y * y + z * z);
    lds_colmin[i] = 0xFFFFFFFFu;
  }
  if (tid < 128) {
    float x = P1[(nbase + tid) * 3 + 0];
    float y = P1[(nbase + tid) * 3 + 1];
    float z = P1[(nbase + tid) * 3 + 2];
    lds_sq1[tid] = x * x + y * y + z * z;
  }
  __syncthreads();

  int wv   = tid >> 5;        // wave in block: 0..7
  int lane = tid & 31;
  int hi   = lane >> 4;       // half-wave: 0 or 1
  int lcol = lane & 15;       // column within 16-wide tile
  int rowbase = nbase + wv * 16;

  // A operand (16x4 f32): lanes 0-15 hold (K0,K1)=(-2x,-2y); lanes 16-31 (K2,K3)=(-2z,0)
  int arow = rowbase + lcol;
  float ax = P1[arow * 3 + 0];
  float ay = P1[arow * 3 + 1];
  float az = P1[arow * 3 + 2];
  v2f A;
  A.x = hi ? (-2.0f * az) : (-2.0f * ax);
  A.y = hi ? 0.0f         : (-2.0f * ay);

  // sq1 per accumulator register: VGPR k <-> row (k + 8*hi)
  v8f s1, rowmin;
#pragma unroll
  for (int k = 0; k < 8; ++k) {
    s1[k] = lds_sq1[wv * 16 + 8 * hi + k];
    rowmin[k] = 3.0e38f;
  }

#pragma unroll 2
  for (int mt = 0; mt < MT_PER_BLOCK; ++mt) {
    int mc = mt * 16 + lcol;
    float4 q = lds_p2q[mc];   // single ds_load_b128: x, y, z, sq2

    // B operand (4x16 f32): lanes 0-15 (K0,K1)=(x,y); lanes 16-31 (K2,K3)=(z,0)
    v2f Bm;
    Bm.x = hi ? q.z : q.x;
    Bm.y = hi ? 0.0f : q.y;

    v8f C;
#pragma unroll
    for (int k = 0; k < 8; ++k) C[k] = s1[k] + q.w;

    // D[row][col] = sq1 + sq2 - 2*<p1,p2>
    v8f D = __builtin_amdgcn_wmma_f32_16x16x4_f32(
        false, A, false, Bm, (short)0, C, false, false);

    float cm = D[0];
#pragma unroll
    for (int k = 0; k < 8; ++k) {
      rowmin[k] = fminf(rowmin[k], D[k]);
      if (k) cm = fminf(cm, D[k]);
    }
    // Both half-waves hit the same LDS slot: ds_min_u32 merges the two
    // 8-row partials -> full 16-row column min. No divergence, EXEC stays all-1s.
    atomicMin(&lds_colmin[mc], encMinF(cm));
  }

  // Row minima: reduce across the 16 lanes of each half-wave, then one atomic each
#pragma unroll
  for (int off = 1; off < 16; off <<= 1) {
#pragma unroll
    for (int k = 0; k < 8; ++k)
      rowmin[k] = fminf(rowmin[k], __shfl_xor(rowmin[k], off, 32));
  }
  if (lcol == 0) {
#pragma unroll
    for (int k = 0; k < 8; ++k)
      atomicMin(&min_n[(size_t)b * PTS + rowbase + 8 * hi + k],
                encMinF(rowmin[k]));
  }

  // Flush block-local column minima to global (1/8 the atomic traffic of per-tile)
  __syncthreads();
  for (int i = tid; i < MCOLS; i += 256)
    atomicMin(&min_m[(size_t)b * PTS + mstart + i], lds_colmin[i]);
}

// One block per (batch, direction): logsumexp(ALPHA*x)/ALPHA over 4096 encoded values.
__global__ __launch_bounds__(256) void hd_lse(const unsigned* __restrict__ mins,
                                              float* __restrict__ partials) {
  __shared__ float red[256];
  int base = blockIdx.x * PTS;
  int tid = threadIdx.x;

  float m = -3.0e38f;
  for (int i = tid; i < PTS; i += 256) m = fmaxf(m, decMinF(mins[base + i]));
  red[tid] = m;
  __syncthreads();
  for (int s = 128; s > 0; s >>= 1) {
    if (tid < s) red[tid] = fmaxf(red[tid], red[tid + s]);
    __syncthreads();
  }
  float mx = red[0];
  __syncthreads();

  float sum = 0.0f;
  for (int i = tid; i < PTS; i += 256)
    sum += expf(ALPHAF * (decMinF(mins[base + i]) - mx));
  red[tid] = sum;
  __syncthreads();
  for (int s = 128; s > 0; s >>= 1) {
    if (tid < s) red[tid] += red[tid + s];
    __syncthreads();
  }
  if (tid == 0) partials[blockIdx.x] = mx + logf(red[0]) / ALPHAF;
}

__global__ void hd_final(const float* __restrict__ partials, float* __restrict__ out) {
  if (threadIdx.x == 0) {
    float s = 0.0f;
    for (int i = 0; i < 2 * BATCH; ++i) s += partials[i];
    // (mean over 4 fwd + mean over 4 bwd)/2 == sum/8
    out[0] = s * 0.125f;
  }
}

extern "C" void kernel_launch(void* const* d_in, const int* in_sizes, int n_in,
                              void* d_out, int out_size, void* d_ws, size_t ws_size,
                              hipStream_t stream) {
  const float* p1 = (const float*)d_in[0];
  const float* p2 = (const float*)d_in[1];
  float* out = (float*)d_out;

  unsigned* min_n    = (unsigned*)d_ws;            // [4][4096] encoded
  unsigned* min_m    = min_n + BATCH * PTS;        // [4][4096] encoded
  float*    partials = (float*)(min_m + BATCH * PTS); // [8]

  int initN = 2 * BATCH * PTS;
  hd_init<<<(initN + 255) / 256, 256, 0, stream>>>(min_n, initN);

  int nBlocks = BATCH * (PTS / 128) * MSPLIT; // 256 blocks of 8 waves
  hd_main<<<nBlocks, 256, 0, stream>>>(p1, p2, min_n, min_m);

  hd_lse<<<2 * BATCH, 256, 0, stream>>>(min_n, partials);
  hd_final<<<1, 32, 0, stream>>>(partials, out);
}